// RCGNLayer_61400852463646
// MI455X (gfx1250) — compile-verified
//
#include <hip/hip_runtime.h>

typedef float v2f __attribute__((ext_vector_type(2)));
typedef float v8f __attribute__((ext_vector_type(8)));

#define D 128   // D_IN == D_OUT

// ---------------------------------------------------------------------------
// Zero a float4 buffer (grid-stride)
// ---------------------------------------------------------------------------
__global__ __launch_bounds__(256)
void rgcn_zero_f4(float4* __restrict__ p, long long n4) {
    long long i = (long long)blockIdx.x * 256 + threadIdx.x;
    long long stride = (long long)gridDim.x * 256;
    for (; i < n4; i += stride) p[i] = make_float4(0.f, 0.f, 0.f, 0.f);
}

// ---------------------------------------------------------------------------
// Transpose W0 and all W_r into Wt ([(R+1)][N=128][K=128], Wt[n][k] = W[k][n])
// so GEMM B-fragments are contiguous in K. Wt[0] = W0^T, Wt[1+r] = W_r^T.
// ---------------------------------------------------------------------------
__global__ __launch_bounds__(256)
void rgcn_transpose_w(const float* __restrict__ W,   // [R,128,128]
                      const float* __restrict__ W0,  // [128,128]
                      float* __restrict__ Wt,        // [(R+1),128,128]
                      int total)                     // (R+1)*16384
{
    int idx = blockIdx.x * 256 + threadIdx.x;
    if (idx >= total) return;
    int mat = idx >> 14;
    int e   = idx & 16383;
    int k   = e >> 7;
    int n   = e & 127;
    const float* src = (mat == 0) ? W0 : (W + (size_t)(mat - 1) * D * D);
    Wt[(size_t)mat * D * D + (size_t)n * D + k] = src[(size_t)k * D + n];
}

// ---------------------------------------------------------------------------
// Edge scatter: H[(rel,) dst] += X[src].
//   relFilter == -1 : single pass over all relations, H is [R][Nn][128]
//   relFilter >= 0  : only edges of that relation, H is [Nn][128]
// Each wave loads a coalesced batch of 32 edge records, then broadcasts one
// edge at a time (__shfl) to the full wave: 32-lane gather of the source row
// (float4/lane) + 4 global f32 atomic adds per lane into the dst row.
// ---------------------------------------------------------------------------
__global__ __launch_bounds__(256)
void rgcn_scatter(const float4* __restrict__ X4,   // X viewed as [Nn][32] float4
                  const int* __restrict__ esrc,
                  const int* __restrict__ edst,
                  const int* __restrict__ erel,
                  float* __restrict__ H,
                  int E, int Nn, int relFilter)
{
    const int lane = threadIdx.x & 31;
    const int w    = blockIdx.x * 8 + (threadIdx.x >> 5);
    const int nw   = gridDim.x * 8;
    for (long long base = (long long)w * 32; base < E; base += (long long)nw * 32) {
        long long idx = base + lane;
        if (idx >= E) idx = E - 1;                   // clamp; tail bounded by cnt
        const int er = erel[idx];
        const int es = esrc[idx];
        const int ed = edst[idx];
        int cnt = (int)(E - base);
        if (cnt > 32) cnt = 32;
        for (int i = 0; i < cnt; ++i) {
            const int r = __shfl(er, i);
            if (relFilter >= 0 && r != relFilter) continue;
            const int s = __shfl(es, i);
            const int d = __shfl(ed, i);
            float4 v = X4[(size_t)s * 32 + lane];
            size_t off = (relFilter >= 0) ? 0 : (size_t)r * Nn * D;
            float* hd = H + off + (size_t)d * D + lane * 4;
            atomicAdd(hd + 0, v.x);
            atomicAdd(hd + 1, v.y);
            atomicAdd(hd + 2, v.z);
            atomicAdd(hd + 3, v.w);
        }
    }
}

// ---------------------------------------------------------------------------
// One 128-wide K segment of the GEMM, accumulated into the wave's 4x4 grid of
// 16x16 f32 WMMA accumulators (K stepped by 4 via V_WMMA_F32_16X16X4_F32).
// A-fragment layout: lanes 0-15 hold K=0,1 of A-rows 0..15; lanes 16-31 hold
// K=2,3 (32-bit A 16x4 layout, ISA 7.12.2). B mirrors with N across lanes.
// Row validity and the optional per-row scale are folded into the A fragment
// so EXEC stays all-ones through the WMMA region (ISA requirement).
// ---------------------------------------------------------------------------
__device__ __forceinline__
void gemm_segment(const float* __restrict__ Aseg,     // [M,128]
                  const float* __restrict__ Wts,      // [128(N),128(K)] (W^T)
                  const float* __restrict__ rowscale, // [M] or nullptr
                  int M, int mbase, int nbase, int lidx, int khalf,
                  v8f acc[4][4])
{
    const float* aptr[4];
    float scl[4];
#pragma unroll
    for (int mi = 0; mi < 4; ++mi) {
        int r   = mbase + mi * 16 + lidx;
        bool ok = (r < M);
        int rr  = ok ? r : 0;
        scl[mi]  = ok ? (rowscale ? rowscale[rr] : 1.0f) : 0.0f;
        aptr[mi] = Aseg + (size_t)rr * D;
    }
    const float* bptr[4];
#pragma unroll
    for (int ni = 0; ni < 4; ++ni)
        bptr[ni] = Wts + (size_t)(nbase + ni * 16 + lidx) * D;

    for (int kk = 0; kk < D / 4; ++kk) {
        const int c0 = 4 * kk + 2 * khalf;
        v2f a[4], b[4];
#pragma unroll
        for (int mi = 0; mi < 4; ++mi) {
            float2 t = *(const float2*)(aptr[mi] + c0);
            a[mi] = (v2f){t.x * scl[mi], t.y * scl[mi]};
        }
#pragma unroll
        for (int ni = 0; ni < 4; ++ni) {
            float2 t = *(const float2*)(bptr[ni] + c0);
            b[ni] = (v2f){t.x, t.y};
        }
#pragma unroll
        for (int mi = 0; mi < 4; ++mi)
#pragma unroll
            for (int ni = 0; ni < 4; ++ni)
                acc[mi][ni] = __builtin_amdgcn_wmma_f32_16x16x4_f32(
                    false, a[mi], false, b[ni], (short)0, acc[mi][ni],
                    false, false);
    }
}

// Epilogue: C/D layout is lanes 0-15 -> M=vgpr(0..7), N=lane;
//           lanes 16-31 -> M=8+vgpr, N=lane-16.
__device__ __forceinline__
void gemm_store(float* __restrict__ out, v8f acc[4][4], int M,
                int mbase, int nbase, int lidx, int khalf,
                bool fullM, bool accumulate)
{
#pragma unroll
    for (int mi = 0; mi < 4; ++mi) {
#pragma unroll
        for (int ni = 0; ni < 4; ++ni) {
            const int c  = nbase + ni * 16 + lidx;
            const int r0 = mbase + mi * 16 + khalf * 8;
#pragma unroll
            for (int v = 0; v < 8; ++v) {
                int gr = r0 + v;
                if (fullM || gr < M) {
                    size_t o  = (size_t)gr * D + c;
                    float val = acc[mi][ni][v];
                    if (accumulate) out[o] += val;
                    else            out[o]  = val;
                }
            }
        }
    }
}

// ---------------------------------------------------------------------------
// Fused GEMM over K = (R+1)*128:
//   out = X @ W0 + sum_r (inv_norm_r .* Hall_r) @ W_r     (single write pass)
// Block: 256 threads = 8 waves arranged 4(M) x 2(N); block tile 256x128.
// Wave tile: 64x64 = 4x4 WMMA accumulators held across all segments.
// ---------------------------------------------------------------------------
__global__ __launch_bounds__(256)
void rgcn_gemm_fused(const float* __restrict__ X,        // [M,128]
                     const float* __restrict__ Hall,     // [R,M,128]
                     const float* __restrict__ Wt,       // [(R+1),128,128]
                     const float* __restrict__ inv_norm, // [R,M]
                     float* __restrict__ out,            // [M,128]
                     int M, int R)
{
    const int tid   = threadIdx.x;
    const int wave  = tid >> 5;
    const int lidx  = tid & 15;
    const int khalf = (tid >> 4) & 1;
    const int wm    = wave >> 1;       // 0..3 (M)
    const int wn    = wave & 1;        // 0..1 (N)
    const int mbase = blockIdx.x * 256 + wm * 64;
    const int nbase = wn * 64;

    v8f acc[4][4];
#pragma unroll
    for (int mi = 0; mi < 4; ++mi)
#pragma unroll
        for (int ni = 0; ni < 4; ++ni)
            acc[mi][ni] = (v8f){0.f, 0.f, 0.f, 0.f, 0.f, 0.f, 0.f, 0.f};

    for (int seg = 0; seg <= R; ++seg) {
        const float* Aseg = (seg == 0) ? X
                          : (Hall + (size_t)(seg - 1) * (size_t)M * D);
        const float* scl  = (seg == 0) ? nullptr
                          : (inv_norm + (size_t)(seg - 1) * M);
        gemm_segment(Aseg, Wt + (size_t)seg * D * D, scl,
                     M, mbase, nbase, lidx, khalf, acc);
    }

    const bool fullM = (blockIdx.x * 256 + 256) <= M;
    gemm_store(out, acc, M, mbase, nbase, lidx, khalf, fullM, false);
}

// ---------------------------------------------------------------------------
// Single-segment GEMM (fallback path): out (+)= (scale .* A) @ Wts^T
// ---------------------------------------------------------------------------
__global__ __launch_bounds__(256)
void rgcn_gemm_wmma(const float* __restrict__ A,
                    const float* __restrict__ Wts,
                    const float* __restrict__ rowscale,
                    float* __restrict__ out,
                    int M, int accumulate)
{
    const int tid   = threadIdx.x;
    const int wave  = tid >> 5;
    const int lidx  = tid & 15;
    const int khalf = (tid >> 4) & 1;
    const int wm    = wave >> 1;
    const int wn    = wave & 1;
    const int mbase = blockIdx.x * 256 + wm * 64;
    const int nbase = wn * 64;

    v8f acc[4][4];
#pragma unroll
    for (int mi = 0; mi < 4; ++mi)
#pragma unroll
        for (int ni = 0; ni < 4; ++ni)
            acc[mi][ni] = (v8f){0.f, 0.f, 0.f, 0.f, 0.f, 0.f, 0.f, 0.f};

    gemm_segment(A, Wts, rowscale, M, mbase, nbase, lidx, khalf, acc);

    const bool fullM = (blockIdx.x * 256 + 256) <= M;
    gemm_store(out, acc, M, mbase, nbase, lidx, khalf, fullM, accumulate != 0);
}

// ---------------------------------------------------------------------------
// Host launcher
// ---------------------------------------------------------------------------
extern "C" void kernel_launch(void* const* d_in, const int* in_sizes, int n_in,
                              void* d_out, int out_size, void* d_ws, size_t ws_size,
                              hipStream_t stream) {
    const float* X        = (const float*)d_in[0];
    const float* W        = (const float*)d_in[1];
    const float* W0       = (const float*)d_in[2];
    const float* inv_norm = (const float*)d_in[3];
    const int*   esrc     = (const int*)d_in[4];
    const int*   edst     = (const int*)d_in[5];
    const int*   erel     = (const int*)d_in[6];
    float*       out      = (float*)d_out;

    const int Nn = in_sizes[0] / D;           // 50000 nodes
    const int R  = in_sizes[1] / (D * D);     // 8 relations
    const int E  = in_sizes[4];               // 800000 edges

    const int wtTotal   = (R + 1) * D * D;
    const int gemmGrid  = (Nn + 255) / 256;
    const size_t wtElems = (size_t)wtTotal;
    const size_t hallElems = (size_t)R * Nn * D;
    const size_t needFused = (hallElems + wtElems) * sizeof(float);

    if (ws_size >= needFused) {
        // --- Fused path: H_all for all relations + single K=(R+1)*128 GEMM ---
        float* Hall = (float*)d_ws;
        float* Wt   = Hall + hallElems;

        rgcn_transpose_w<<<(wtTotal + 255) / 256, 256, 0, stream>>>(W, W0, Wt, wtTotal);

        long long n4 = (long long)hallElems / 4;
        rgcn_zero_f4<<<4096, 256, 0, stream>>>((float4*)Hall, n4);

        rgcn_scatter<<<2048, 256, 0, stream>>>((const float4*)X, esrc, edst, erel,
                                               Hall, E, Nn, /*relFilter=*/-1);

        rgcn_gemm_fused<<<gemmGrid, 256, 0, stream>>>(X, Hall, Wt, inv_norm,
                                                      out, Nn, R);
    } else {
        // --- Fallback: one relation at a time (needs only ~26 MB workspace) ---
        float* H  = (float*)d_ws;
        float* Wt = H + (size_t)Nn * D;

        rgcn_transpose_w<<<(wtTotal + 255) / 256, 256, 0, stream>>>(W, W0, Wt, wtTotal);

        rgcn_gemm_wmma<<<gemmGrid, 256, 0, stream>>>(X, Wt, nullptr, out, Nn, 0);

        long long n4 = (long long)Nn * D / 4;
        for (int r = 0; r < R; ++r) {
            rgcn_zero_f4<<<2048, 256, 0, stream>>>((float4*)H, n4);
            rgcn_scatter<<<2048, 256, 0, stream>>>((const float4*)X, esrc, edst, erel,
                                                   H, E, Nn, r);
            rgcn_gemm_wmma<<<gemmGrid, 256, 0, stream>>>(
                H, Wt + (size_t)(1 + r) * D * D, inv_norm + (size_t)r * Nn,
                out, Nn, 1);
        }
    }
}